// LocalTrans_75136157876252
// MI455X (gfx1250) — compile-verified
//
#include <hip/hip_runtime.h>
#include <hip/hip_bf16.h>

// Problem constants (fixed by the reference).
#define BB   4
#define NN   16384
#define KK   16
#define CC   64
#define BN   (BB * NN)        // 65536 rows
#define NEG_SLOPE 0.2f
#define EPS_BN    1e-5f

typedef __attribute__((ext_vector_type(2))) float v2f;
typedef __attribute__((ext_vector_type(8))) float v8f;

// ---------------------------------------------------------------------------
// Kernel 0: zero the stats buffer (sum[64] + sumsq[64]) — ws is poisoned.
// ---------------------------------------------------------------------------
__global__ void zero_stats_kernel(float* __restrict__ stats) {
    if (threadIdx.x < 128) stats[threadIdx.x] = 0.0f;
}

// ---------------------------------------------------------------------------
// Kernels 1 & 3: fused multi-matrix GEMM.
//   O_g[M,64] = A[M,64] @ W_g[64,64] + bias_g    for g in [0, nmat)
// fp32 via V_WMMA_F32_16X16X4_F32.  256 threads = 8 wave32; each wave owns a
// 16-row x 64-col strip (4 tiles of 16x16), accumulating K=64 in steps of 4.
//
// LDS layouts are chosen so every WMMA fragment is ONE aligned ds_load_b64:
//   lA : 128 rows x 64 K, padded to stride 66 (even => 8B-aligned pairs)
//   lW : paired-K layout, element (k,n) at (k>>1)*128 + 2n + (k&1), so the
//        {K=ko, K=ko+1} B-fragment for column n is contiguous (ko is even).
//
// If `stats` != nullptr (FFN call, nmat==1), also accumulates per-column sum
// and sum-of-squares: ds_add_f32 LDS reduction + one global atomic per column.
// ---------------------------------------------------------------------------
__global__ __launch_bounds__(256)
void gemm64_kernel(const float* __restrict__ A,
                   const float* __restrict__ W0, const float* __restrict__ W1,
                   const float* __restrict__ W2,
                   const float* __restrict__ b0, const float* __restrict__ b1,
                   const float* __restrict__ b2,
                   float* __restrict__ O0, float* __restrict__ O1,
                   float* __restrict__ O2,
                   int nmat, float* __restrict__ stats) {
    __shared__ float lA[128 * 66];      // 33792 B
    __shared__ float lW[3 * 64 * 64];   // 49152 B, paired-K layout
    __shared__ float lstat[128];

    const int t    = threadIdx.x;
    const int lane = t & 31;
    const int wv   = t >> 5;            // wave id 0..7
    const int m    = lane & 15;         // A row-within-tile / B,C,D column
    const int hi   = lane >> 4;         // half-wave select

    const int rowBase = blockIdx.x * 128;

    // Stage the 128x64 A tile (coalesced, stride-66 padded).
    #pragma unroll
    for (int i = 0; i < 32; ++i) {
        const int j   = t + 256 * i;
        const int row = j >> 6;
        const int col = j & 63;
        lA[row * 66 + col] = A[(size_t)(rowBase + row) * CC + col];
    }
    // Stage weight matrices in paired-K layout.
    for (int g = 0; g < nmat; ++g) {
        const float* Wg = (g == 0) ? W0 : ((g == 1) ? W1 : W2);
        #pragma unroll
        for (int i = 0; i < 16; ++i) {
            const int j = t + 256 * i;
            const int k = j >> 6;
            const int n = j & 63;
            lW[g * 4096 + (k >> 1) * 128 + 2 * n + (k & 1)] = Wg[j];
        }
    }
    if (stats && t < 128) lstat[t] = 0.0f;
    __syncthreads();

    const int rowW = wv * 16;           // this wave's 16-row strip

    for (int g = 0; g < nmat; ++g) {
        const float* biasg = (g == 0) ? b0 : ((g == 1) ? b1 : b2);
        float*       Og    = (g == 0) ? O0 : ((g == 1) ? O1 : O2);

        // Accumulators seeded with bias (broadcast along M).
        v8f acc[4];
        #pragma unroll
        for (int ct = 0; ct < 4; ++ct) {
            const float bb = biasg[ct * 16 + m];
            #pragma unroll
            for (int j = 0; j < 8; ++j) acc[ct][j] = bb;
        }

        const float* lWg = &lW[g * 4096];
        for (int kk = 0; kk < 64; kk += 4) {
            const int ko = kk + hi * 2;   // even: lanes<16 K={kk,kk+1}, >=16 {kk+2,kk+3}
            const v2f afrag =
                *reinterpret_cast<const v2f*>(&lA[(rowW + m) * 66 + ko]);
            const int brow = (ko >> 1) * 128;
            #pragma unroll
            for (int ct = 0; ct < 4; ++ct) {
                const v2f bfrag = *reinterpret_cast<const v2f*>(
                    &lWg[brow + 2 * (ct * 16 + m)]);
                acc[ct] = __builtin_amdgcn_wmma_f32_16x16x4_f32(
                    false, afrag, false, bfrag, (short)0, acc[ct], false, false);
            }
        }

        // Store D: VGPR j -> row (rowW + hi*8 + j), column (ct*16 + m).
        const int outRow = rowBase + rowW + hi * 8;
        #pragma unroll
        for (int ct = 0; ct < 4; ++ct) {
            #pragma unroll
            for (int j = 0; j < 8; ++j) {
                Og[(size_t)(outRow + j) * CC + ct * 16 + m] = acc[ct][j];
            }
        }

        // Per-column training-stat accumulation (BatchNorm), FFN call only.
        if (stats) {
            #pragma unroll
            for (int ct = 0; ct < 4; ++ct) {
                float ps = 0.0f, pq = 0.0f;
                #pragma unroll
                for (int j = 0; j < 8; ++j) {
                    const float d = acc[ct][j];
                    ps += d;
                    pq += d * d;
                }
                atomicAdd(&lstat[ct * 16 + m], ps);       // ds_add_f32
                atomicAdd(&lstat[64 + ct * 16 + m], pq);
            }
            __syncthreads();
            if (t < 128) atomicAdd(&stats[t], lstat[t]);  // global_atomic_add_f32
        }
    }
}

// ---------------------------------------------------------------------------
// Kernel 2: subtraction-based local attention. One wave32 per point; each
// lane owns channels {2*lane, 2*lane+1} (coalesced float2 accesses).
//   e[k]   = (q - k_gathered[k]) / sqrt(C)
//   attn   = softmax_k(e) - 1            (sum over k of softmax == 1)
//   ctx[c] = max_k(attn[k,c] * v_gathered[k,c])
// ---------------------------------------------------------------------------
__global__ __launch_bounds__(256)
void attn64_kernel(const float* __restrict__ qbuf,
                   const float* __restrict__ kbuf,
                   const float* __restrict__ vbuf,
                   const int* __restrict__ idx,
                   float* __restrict__ ctx) {
    const int lane = threadIdx.x & 31;
    const int wv   = threadIdx.x >> 5;
    const int p    = blockIdx.x * 8 + wv;       // point id in [0, BN)
    const int brow = (p >> 14) << 14;           // batch base row = (p/N)*N

    const float2 qq =
        reinterpret_cast<const float2*>(qbuf + (size_t)p * CC)[lane];

    const int ibase = p * KK;
    float ex[KK], ey[KK];
    float mx = -1e30f, my = -1e30f;

    #pragma unroll
    for (int k = 0; k < KK; ++k) {
        const int row = brow + idx[ibase + k];
        const float2 kv =
            reinterpret_cast<const float2*>(kbuf + (size_t)row * CC)[lane];
        ex[k] = (qq.x - kv.x) * 0.125f;         // / sqrt(64)
        ey[k] = (qq.y - kv.y) * 0.125f;
        mx = fmaxf(mx, ex[k]);
        my = fmaxf(my, ey[k]);
    }

    float sx = 0.0f, sy = 0.0f;
    #pragma unroll
    for (int k = 0; k < KK; ++k) {
        ex[k] = __expf(ex[k] - mx);
        ey[k] = __expf(ey[k] - my);
        sx += ex[k];
        sy += ey[k];
    }
    const float ivx = 1.0f / sx;
    const float ivy = 1.0f / sy;

    float cx = -1e30f, cy = -1e30f;
    #pragma unroll
    for (int k = 0; k < KK; ++k) {
        const int row = brow + idx[ibase + k];
        const float2 vv =
            reinterpret_cast<const float2*>(vbuf + (size_t)row * CC)[lane];
        const float ax = ex[k] * ivx - 1.0f;    // softmax offset-subtract
        const float ay = ey[k] * ivy - 1.0f;
        cx = fmaxf(cx, ax * vv.x);
        cy = fmaxf(cy, ay * vv.y);
    }

    float2 r; r.x = cx; r.y = cy;
    reinterpret_cast<float2*>(ctx + (size_t)p * CC)[lane] = r;
}

// ---------------------------------------------------------------------------
// Kernel 4: BatchNorm (training stats) + LeakyReLU(0.2) + residual.
// ---------------------------------------------------------------------------
__global__ __launch_bounds__(256)
void finalize_kernel(const float* __restrict__ feats,
                     const float* __restrict__ h,
                     const float* __restrict__ stats,
                     const float* __restrict__ gamma,
                     const float* __restrict__ beta,
                     float* __restrict__ out) {
    const int gid = blockIdx.x * 256 + threadIdx.x;   // < BN*CC = 4194304
    const int c   = gid & 63;
    const float invBN = 1.0f / (float)BN;
    const float mean = stats[c] * invBN;
    const float var  = stats[64 + c] * invBN - mean * mean;
    const float rs   = rsqrtf(var + EPS_BN);
    float y = (h[gid] - mean) * rs * gamma[c] + beta[c];
    y = (y >= 0.0f) ? y : NEG_SLOPE * y;
    out[gid] = feats[gid] + y;
}

// ---------------------------------------------------------------------------
extern "C" void kernel_launch(void* const* d_in, const int* in_sizes, int n_in,
                              void* d_out, int out_size, void* d_ws, size_t ws_size,
                              hipStream_t stream) {
    const float* features = (const float*)d_in[0];
    // d_in[1] = pos (unused by the reference math)
    const float* qw    = (const float*)d_in[2];
    const float* qb    = (const float*)d_in[3];
    const float* kw    = (const float*)d_in[4];
    const float* kb    = (const float*)d_in[5];
    const float* vw    = (const float*)d_in[6];
    const float* vb    = (const float*)d_in[7];
    const float* fw    = (const float*)d_in[8];
    const float* fb    = (const float*)d_in[9];
    const float* gamma = (const float*)d_in[10];
    const float* beta  = (const float*)d_in[11];
    const int*   idx   = (const int*)d_in[12];
    float* out = (float*)d_out;

    const size_t mat = (size_t)BN * CC;       // 4,194,304 floats (16 MB)
    float* qbuf  = (float*)d_ws;
    float* kbuf  = qbuf + mat;
    float* vbuf  = kbuf + mat;
    float* ctx   = vbuf + mat;
    float* stats = ctx + mat;                 // 128 floats
    float* hbuf  = qbuf;                      // q is dead after attention

    zero_stats_kernel<<<1, 128, 0, stream>>>(stats);

    // Fused q/k/v projections: one pass over features, three weight matrices.
    gemm64_kernel<<<BN / 128, 256, 0, stream>>>(
        features, qw, kw, vw, qb, kb, vb, qbuf, kbuf, vbuf, 3, nullptr);

    // gather + softmax-offset attention + neighbor max
    attn64_kernel<<<BN / 8, 256, 0, stream>>>(qbuf, kbuf, vbuf, idx, ctx);

    // FFN GEMM + per-channel sum/sumsq for BatchNorm training stats
    gemm64_kernel<<<BN / 128, 256, 0, stream>>>(
        ctx, fw, fw, fw, fb, fb, fb, hbuf, hbuf, hbuf, 1, stats);

    // BN + LeakyReLU + residual
    finalize_kernel<<<(BN * CC) / 256, 256, 0, stream>>>(
        features, hbuf, stats, gamma, beta, out);
}